// Attention_46256797778323
// MI455X (gfx1250) — compile-verified
//
#include <hip/hip_runtime.h>
#include <math.h>

// Problem constants (from reference): B=64, S=4096, T=1024, fp32.
#define B_DIM 64
#define S_DIM 4096
#define T_DIM 1024
#define SPLITS 16                             // S-splits per batch (parallelism)
#define WAVES  8                              // waves per block in flash kernel
#define ROWS_PER_SPLIT (S_DIM / SPLITS)       // 256
#define ROWS_PER_WAVE  (ROWS_PER_SPLIT / WAVES) // 32 contiguous rows per wave

typedef __attribute__((ext_vector_type(2))) float v2f;
typedef __attribute__((ext_vector_type(4))) float v4f;   // native vector: OK for NT loads
typedef __attribute__((ext_vector_type(8))) float v8f;

// ---------------------------------------------------------------------------
// Kernel 1: q = tanh(t[:,idx,:] @ W^T + bias)   via V_WMMA_F32_16X16X4_F32
// GEMM M=64 (batch), N=1024, K=1024.  One wave per 16x16 output tile.
// A-frag layout (ISA 7.12.2, 32-bit A 16x4): lanes 0-15 hold K=0,1; lanes
// 16-31 hold K=2,3; M = lane%16.  B-frag mirrors with lane holding column N.
// B[k][n] = W[n][k]  -> lane n reads W row n contiguously along k.
// C/D layout: VGPR v -> M = v + (lane<16 ? 0 : 8), N = lane%16.
// ---------------------------------------------------------------------------
__global__ __launch_bounds__(32)
void qgemm_wmma(const int* __restrict__ idxp,
                const float* __restrict__ t,
                const float* __restrict__ W,
                const float* __restrict__ bias,
                float* __restrict__ q) {
    const int lane = threadIdx.x;
    const int half = lane >> 4;      // 0: K=0,1   1: K=2,3
    const int lm   = lane & 15;
    const int n0   = blockIdx.x * 16;
    const int m0   = blockIdx.y * 16;
    const int idx  = *idxp;

    const float* Arow = t + ((size_t)(m0 + lm) * S_DIM + (size_t)idx) * T_DIM;
    const float* Brow = W + (size_t)(n0 + lm) * T_DIM;   // column n of B = row n of W

    v8f c = {0.f, 0.f, 0.f, 0.f, 0.f, 0.f, 0.f, 0.f};
    for (int k0 = 0; k0 < T_DIM; k0 += 4) {
        const int kb = k0 + half * 2;
        const v2f a = *(const v2f*)(Arow + kb);
        const v2f b = *(const v2f*)(Brow + kb);
        c = __builtin_amdgcn_wmma_f32_16x16x4_f32(
                /*neg_a=*/false, a, /*neg_b=*/false, b,
                /*c_mod=*/(short)0, c, /*reuse_a=*/false, /*reuse_b=*/false);
    }
#pragma unroll
    for (int v = 0; v < 8; ++v) {
        const int row = m0 + v + half * 8;
        const int col = n0 + lm;
        q[row * T_DIM + col] = tanhf(c[v] + bias[col]);
    }
}

// ---------------------------------------------------------------------------
// Kernel 2: fused one-pass online-softmax attention (flash style).
// grid = (SPLITS, B), block = 256 (8 waves).  Each wave streams 32 contiguous
// rows (4 KB each); lane l owns elements {i*128 + l*4 .. +3 | i<8} of the
// 1024-dim vectors.  Running (m, l, acc[32 floats/lane]) kept in VGPRs.
// t_matrix is read exactly once (NT hint) -> ~1 GiB total HBM traffic.
// ---------------------------------------------------------------------------
__global__ __launch_bounds__(256)
void flash_pass(const float* __restrict__ t,
                const float* __restrict__ q,
                float* __restrict__ pm,
                float* __restrict__ pl,
                float* __restrict__ pacc) {
    const int tid   = threadIdx.x;
    const int lane  = tid & 31;
    const int wave  = tid >> 5;
    const int split = blockIdx.x;
    const int b     = blockIdx.y;
    const int srow0 = split * ROWS_PER_SPLIT + wave * ROWS_PER_WAVE;

    // q fragment for this batch (same element mapping as rows)
    const v4f* qb = (const v4f*)(q + (size_t)b * T_DIM);
    v4f qf[8];
#pragma unroll
    for (int i = 0; i < 8; ++i) qf[i] = qb[i * 32 + lane];

    v4f acc[8];
#pragma unroll
    for (int i = 0; i < 8; ++i) acc[i] = (v4f){0.f, 0.f, 0.f, 0.f};
    float m = -INFINITY;
    float l = 0.f;

    const float* base = t + ((size_t)b * S_DIM + (size_t)srow0) * T_DIM;
    for (int r = 0; r < ROWS_PER_WAVE; ++r) {
        const v4f* row = (const v4f*)(base + (size_t)r * T_DIM);
        v4f rf[8];
#pragma unroll
        for (int i = 0; i < 8; ++i) {
            rf[i] = __builtin_nontemporal_load(&row[i * 32 + lane]);
            // prefetch same chunk of the next row (sequential 128 KB stream
            // per wave); speculative, safe if it runs past the end.
            __builtin_prefetch((const void*)(row + 256 + i * 32 + lane), 0, 0);
        }
        // lane-partial dot(q, row)
        float z = 0.f;
#pragma unroll
        for (int i = 0; i < 8; ++i) {
            z = fmaf(rf[i].x, qf[i].x, z);
            z = fmaf(rf[i].y, qf[i].y, z);
            z = fmaf(rf[i].z, qf[i].z, z);
            z = fmaf(rf[i].w, qf[i].w, z);
        }
        // wave32 butterfly reduce -> all lanes hold full dot
#pragma unroll
        for (int off = 16; off > 0; off >>= 1) z += __shfl_xor(z, off, 32);

        // online softmax update
        const float mn   = fmaxf(m, z);
        const float corr = __expf(m - mn);   // 1.0 when max unchanged, 0 on first row
        const float p    = __expf(z - mn);
        l = l * corr + p;
#pragma unroll
        for (int i = 0; i < 8; ++i) {
            acc[i].x = fmaf(acc[i].x, corr, p * rf[i].x);
            acc[i].y = fmaf(acc[i].y, corr, p * rf[i].y);
            acc[i].z = fmaf(acc[i].z, corr, p * rf[i].z);
            acc[i].w = fmaf(acc[i].w, corr, p * rf[i].w);
        }
        m = mn;
    }

    // ---- combine the 8 waves of this block in LDS ----
    __shared__ float s_m[WAVES];
    __shared__ float s_l[WAVES];
    __shared__ float sacc[T_DIM];
    ((v4f*)sacc)[tid] = (v4f){0.f, 0.f, 0.f, 0.f};  // 256*16B = 4KB
    if (lane == 0) { s_m[wave] = m; s_l[wave] = l; }
    __syncthreads();

    float M = s_m[0];
#pragma unroll
    for (int w = 1; w < WAVES; ++w) M = fmaxf(M, s_m[w]);
    float Lt = 0.f;
#pragma unroll
    for (int w = 0; w < WAVES; ++w) Lt += s_l[w] * __expf(s_m[w] - M);
    const float fw = __expf(m - M);          // uniform within wave

#pragma unroll
    for (int i = 0; i < 8; ++i) {            // ds_add_f32, conflict-free per wave
        const int e = i * 128 + lane * 4;
        atomicAdd(&sacc[e + 0], acc[i].x * fw);
        atomicAdd(&sacc[e + 1], acc[i].y * fw);
        atomicAdd(&sacc[e + 2], acc[i].z * fw);
        atomicAdd(&sacc[e + 3], acc[i].w * fw);
    }
    __syncthreads();

    const int pidx = b * SPLITS + split;
    if (tid == 0) { pm[pidx] = M; pl[pidx] = Lt; }
    ((v4f*)(pacc + (size_t)pidx * T_DIM))[tid] = ((v4f*)sacc)[tid];
}

// ---------------------------------------------------------------------------
// Kernel 3: combine the SPLITS partials per batch -> t_att[b, :]
// grid = B, block = 256 (each thread owns 4 output elements).
// ---------------------------------------------------------------------------
__global__ __launch_bounds__(256)
void combine_splits(const float* __restrict__ pm,
                    const float* __restrict__ pl,
                    const float* __restrict__ pacc,
                    float* __restrict__ out) {
    const int b   = blockIdx.x;
    const int tid = threadIdx.x;

    float M = -INFINITY;
#pragma unroll
    for (int i = 0; i < SPLITS; ++i) M = fmaxf(M, pm[b * SPLITS + i]);
    float f[SPLITS];
    float L = 0.f;
#pragma unroll
    for (int i = 0; i < SPLITS; ++i) {
        f[i] = __expf(pm[b * SPLITS + i] - M);
        L = fmaf(pl[b * SPLITS + i], f[i], L);
    }
    const float invL = 1.f / L;

    v4f a = (v4f){0.f, 0.f, 0.f, 0.f};
#pragma unroll
    for (int i = 0; i < SPLITS; ++i) {
        const v4f v = ((const v4f*)(pacc + (size_t)(b * SPLITS + i) * T_DIM))[tid];
        a.x = fmaf(v.x, f[i], a.x);
        a.y = fmaf(v.y, f[i], a.y);
        a.z = fmaf(v.z, f[i], a.z);
        a.w = fmaf(v.w, f[i], a.w);
    }
    a.x *= invL; a.y *= invL; a.z *= invL; a.w *= invL;
    ((v4f*)(out + (size_t)b * T_DIM))[tid] = a;
}

// ---------------------------------------------------------------------------
// Launch: inputs are {index:int, t_matrix:(B,S,T) f32, W:(T,T) f32, b:(T,) f32}
// Output: t_att (B,T) f32.  Workspace: q (256KB) + partials (~4.2MB).
// ---------------------------------------------------------------------------
extern "C" void kernel_launch(void* const* d_in, const int* in_sizes, int n_in,
                              void* d_out, int out_size, void* d_ws, size_t ws_size,
                              hipStream_t stream) {
    const int*   idxp = (const int*)d_in[0];
    const float* t    = (const float*)d_in[1];
    const float* W    = (const float*)d_in[2];
    const float* bias = (const float*)d_in[3];
    float*       out  = (float*)d_out;

    float* q    = (float*)d_ws;                 // B*T            = 65536 floats
    float* pm   = q  + (size_t)B_DIM * T_DIM;   // B*SPLITS       = 1024 floats
    float* pl   = pm + (size_t)B_DIM * SPLITS;  // B*SPLITS       = 1024 floats
    float* pacc = pl + (size_t)B_DIM * SPLITS;  // B*SPLITS*T     = 1M floats

    qgemm_wmma<<<dim3(T_DIM / 16, B_DIM / 16), 32, 0, stream>>>(idxp, t, W, bias, q);
    flash_pass<<<dim3(SPLITS, B_DIM), 256, 0, stream>>>(t, q, pm, pl, pacc);
    combine_splits<<<B_DIM, 256, 0, stream>>>(pm, pl, pacc, out);
}